// FilterDetection_15375982920329
// MI455X (gfx1250) — compile-verified
//
#include <hip/hip_runtime.h>
#include <hip/hip_bf16.h>
#include <stdint.h>

// Problem constants (match reference setup_inputs / constants)
#define B_      4
#define N_      200000
#define C_      21
#define FG      20          // foreground classes (class 0 ignored)
#define P_      500         // PERFORMANCE_COUNT
#define K_      100         // PROPOSAL_COUNT
#define OUTD    25          // C + 4
#define NB      256         // histogram buckets over [0,1] (16x16 for WMMA scan)
#define CAP     1024        // candidate cap per (b,class), power of two (bitonic)
#define TILE    256         // anchors per staged logits tile
#define SCORE_THR 0.05f
#define IOU_THR   0.5f
#define MAXRATIO  4.1351665567423560f   // |ln(0.016)|

#if defined(__has_builtin)
#if __has_builtin(__builtin_amdgcn_global_load_async_to_lds_b128)
#define HAVE_ASYNC_LDS 1
#endif
#if __has_builtin(__builtin_amdgcn_wmma_f32_16x16x4_f32)
#define HAVE_WMMA_F32 1
#endif
#endif

// Matches the async builtin's parameter type exactly:
//   'int __attribute__((vector_size(16))) __device__ *'
typedef int v4i __attribute__((vector_size(16)));
#define GPTR(p) ((__attribute__((address_space(1))) v4i*)(p))
#define LPTR(p) ((__attribute__((address_space(3))) v4i*)(p))

typedef float v2f __attribute__((ext_vector_type(2)));
typedef float v8f __attribute__((ext_vector_type(8)));

__device__ __forceinline__ int bucket_of(float s) {
  int b = (int)(s * (float)NB);
  b = b < 0 ? 0 : b;
  return b > (NB - 1) ? (NB - 1) : b;
}

__device__ __forceinline__ float4 decode_box(const float4 a, const float4 r) {
  float w  = a.z - a.x, h = a.w - a.y;
  float cx = a.x + 0.5f * w, cy = a.y + 0.5f * h;
  float dw = fminf(fmaxf(r.z, -MAXRATIO), MAXRATIO);
  float dh = fminf(fmaxf(r.w, -MAXRATIO), MAXRATIO);
  float pcx = cx + r.x * w, pcy = cy + r.y * h;
  float pw = w * expf(dw), ph = h * expf(dh);
  float4 o;
  o.x = fminf(fmaxf(pcx - 0.5f * pw, 0.f), 1.f);
  o.y = fminf(fmaxf(pcy - 0.5f * ph, 0.f), 1.f);
  o.z = fminf(fmaxf(pcx + 0.5f * pw, 0.f), 1.f);
  o.w = fminf(fmaxf(pcy + 0.5f * ph, 0.f), 1.f);
  return o;
}

// Issue an async global->LDS DMA of nfloats contiguous floats (16B-aligned
// base). Does NOT wait — caller pipelines with wait_stage_and_sync().
__device__ __forceinline__ void issue_stage(const float* __restrict__ g,
                                            float* l, int nfloats,
                                            int tid, int nthr) {
  int nchunks = nfloats >> 2;   // 16B chunks
#if defined(HAVE_ASYNC_LDS)
  for (int i = tid; i < nchunks; i += nthr) {
    __builtin_amdgcn_global_load_async_to_lds_b128(
        GPTR(g + (size_t)i * 4), LPTR(l + (size_t)i * 4), 0, 0);
  }
  for (int i = (nchunks << 2) + tid; i < nfloats; i += nthr) l[i] = g[i];
#else
  for (int i = tid; i < nchunks; i += nthr) {
    float4 v = *(const float4*)(g + (size_t)i * 4);
    *(float4*)(l + (size_t)i * 4) = v;
  }
  for (int i = (nchunks << 2) + tid; i < nfloats; i += nthr) l[i] = g[i];
#endif
}

__device__ __forceinline__ void wait_stage_and_sync() {
#if defined(HAVE_ASYNC_LDS)
#if __has_builtin(__builtin_amdgcn_s_wait_asynccnt)
  __builtin_amdgcn_s_wait_asynccnt(0);
#else
  asm volatile("s_wait_asynccnt 0" ::: "memory");
#endif
#endif
  __syncthreads();
}

__global__ void k_zero(int* __restrict__ p, int n) {
  int i = blockIdx.x * blockDim.x + threadIdx.x;
  if (i < n) p[i] = 0;
}

// Pass 1: histogram of s = logits[b,n,c]*score[b,n] for every fg class.
// grid (32, B), block 256. Double-buffered async DMA pipeline:
// LDS = 2 x 21KB tile + 20KB histogram = 62KB.
__global__ __launch_bounds__(256) void k_hist(const float* __restrict__ logits,
                                              const float* __restrict__ score,
                                              int* __restrict__ hist) {
  __shared__ float s_log[2][TILE * C_];
  __shared__ int   s_h[FG * NB];
  const int b = blockIdx.y, tid = threadIdx.x;
  for (int i = tid; i < FG * NB; i += 256) s_h[i] = 0;
  __syncthreads();
  const int ntiles = (N_ + TILE - 1) / TILE;
  const int stride = gridDim.x;
  int t = blockIdx.x;
  if (t < ntiles) {
    const int n0  = t * TILE;
    const int cnt = (N_ - n0) < TILE ? (N_ - n0) : TILE;
    issue_stage(logits + ((size_t)b * N_ + n0) * C_, s_log[0], cnt * C_, tid, 256);
  }
  wait_stage_and_sync();
  int buf = 0;
  for (; t < ntiles; t += stride) {
    const int nt = t + stride;              // prefetch next tile of this block
    if (nt < ntiles) {
      const int n0n  = nt * TILE;
      const int cntn = (N_ - n0n) < TILE ? (N_ - n0n) : TILE;
      issue_stage(logits + ((size_t)b * N_ + n0n) * C_, s_log[buf ^ 1],
                  cntn * C_, tid, 256);
    }
    const int n0  = t * TILE;
    const int cnt = (N_ - n0) < TILE ? (N_ - n0) : TILE;
    if (tid < cnt) {
      const float sc = score[(size_t)b * N_ + n0 + tid];
      #pragma unroll
      for (int c = 1; c < C_; ++c) {
        float s = s_log[buf][tid * C_ + c] * sc;
        atomicAdd(&s_h[(c - 1) * NB + bucket_of(s)], 1);
      }
    }
    wait_stage_and_sync();                  // next tile landed; cur buf free
    buf ^= 1;
  }
  for (int i = tid; i < FG * NB; i += 256) {
    int v = s_h[i];
    if (v) atomicAdd(&hist[(size_t)b * FG * NB + i], v);
  }
}

// Suffix-scan of the 256-bucket histogram via WMMA (one wave per (b,class)):
//   M[r][q] = h[q*16+r]  (column-major 16x16)
//   S = T_u x M + J x (M x T_s)   with T_u[r][r']=(r'>=r), T_s[q'][q]=(q'>q)
// gives S[r][q] = suffix sum over flat index f = q*16+r. Counts <= 200000 are
// exact in f32, so V_WMMA_F32_16X16X4_F32 computes this bit-exactly.
// Threshold = max{f : S[f] >= P}.
__global__ __launch_bounds__(32) void k_thresh(const int* __restrict__ hist,
                                               int* __restrict__ thresh) {
  const int bc = blockIdx.x;
  const int* h = hist + (size_t)bc * NB;
#if defined(HAVE_WMMA_F32)
  const int lane = threadIdx.x;        // 0..31, all active (EXEC all-1s)
  const int r16  = lane & 15;
  const int hi   = lane >> 4;
  // acc = T_u x M   (A = T_u constant, B = M; 4 chained K=4 chunks)
  v8f acc = {};
  #pragma unroll
  for (int c = 0; c < 4; ++c) {
    v2f a{}, bv{};
    #pragma unroll
    for (int v = 0; v < 2; ++v) {
      int k = 4 * c + v + 2 * hi;                  // contraction index
      a[v]  = (k >= r16) ? 1.f : 0.f;              // T_u[M=r16][K=k]
      bv[v] = (float)h[r16 * 16 + k];              // M[K=k][N=r16] = h[N*16+K]
    }
    acc = __builtin_amdgcn_wmma_f32_16x16x4_f32(false, a, false, bv,
                                                (short)0, acc, false, false);
  }
  // q8 = M x T_s    (A = M, B = T_s constant)
  v8f q8 = {};
  #pragma unroll
  for (int c = 0; c < 4; ++c) {
    v2f a{}, bv{};
    #pragma unroll
    for (int v = 0; v < 2; ++v) {
      int k = 4 * c + v + 2 * hi;
      a[v]  = (float)h[k * 16 + r16];              // M[M=r16][K=k] = h[K*16+M]
      bv[v] = (k > r16) ? 1.f : 0.f;               // T_s[K=k][N=r16]
    }
    q8 = __builtin_amdgcn_wmma_f32_16x16x4_f32(false, a, false, bv,
                                               (short)0, q8, false, false);
  }
  // acc += J x q8   (J all-ones is layout-invariant; relayout q8 D->B chunks)
  #pragma unroll
  for (int c = 0; c < 4; ++c) {
    v2f a{}, bv{};
    a[0] = 1.f; a[1] = 1.f;
    #pragma unroll
    for (int v = 0; v < 2; ++v) {
      const int rlo = 4 * c + v;       // row needed by lanes 0-15
      const int rhi = rlo + 2;         // row needed by lanes 16-31
      float blo = __shfl(q8[rlo & 7], ((rlo >> 3) << 4) + r16, 32);
      float bhi = __shfl(q8[rhi & 7], ((rhi >> 3) << 4) + r16, 32);
      bv[v] = hi ? bhi : blo;
    }
    acc = __builtin_amdgcn_wmma_f32_16x16x4_f32(false, a, false, bv,
                                                (short)0, acc, false, false);
  }
  // max flat index with suffix >= P  (D layout: r = v+8*hi, q = r16)
  int m = -1;
  #pragma unroll
  for (int v = 0; v < 8; ++v) {
    int f = r16 * 16 + v + 8 * hi;
    if (acc[v] >= (float)P_) m = (f > m) ? f : m;
  }
  #pragma unroll
  for (int off = 16; off > 0; off >>= 1) {
    int o = __shfl_xor(m, off, 32);
    m = (o > m) ? o : m;
  }
  if (lane == 0) thresh[bc] = m < 0 ? 0 : m;
#else
  if (threadIdx.x != 0) return;
  int acc = 0, t = 0;
  for (int j = NB - 1; j >= 0; --j) {
    acc += h[j];
    if (acc >= P_) { t = j; break; }
  }
  thresh[bc] = t;
#endif
}

// Pass 2: compact candidates with bucket(s) >= threshold bucket.
// Same double-buffered async pipeline; second scan should mostly hit L2
// (logits = 67MB < 192MB L2).
__global__ __launch_bounds__(256) void k_compact(const float* __restrict__ logits,
                                                 const float* __restrict__ score,
                                                 const int* __restrict__ thresh,
                                                 int* __restrict__ cand_cnt,
                                                 float* __restrict__ cand_s,
                                                 int* __restrict__ cand_i) {
  __shared__ float s_log[2][TILE * C_];
  __shared__ int   s_thr[FG];
  const int b = blockIdx.y, tid = threadIdx.x;
  for (int i = tid; i < FG; i += 256) s_thr[i] = thresh[b * FG + i];
  __syncthreads();
  const int ntiles = (N_ + TILE - 1) / TILE;
  const int stride = gridDim.x;
  int t = blockIdx.x;
  if (t < ntiles) {
    const int n0  = t * TILE;
    const int cnt = (N_ - n0) < TILE ? (N_ - n0) : TILE;
    issue_stage(logits + ((size_t)b * N_ + n0) * C_, s_log[0], cnt * C_, tid, 256);
  }
  wait_stage_and_sync();
  int buf = 0;
  for (; t < ntiles; t += stride) {
    const int nt = t + stride;
    if (nt < ntiles) {
      const int n0n  = nt * TILE;
      const int cntn = (N_ - n0n) < TILE ? (N_ - n0n) : TILE;
      issue_stage(logits + ((size_t)b * N_ + n0n) * C_, s_log[buf ^ 1],
                  cntn * C_, tid, 256);
    }
    const int n0  = t * TILE;
    const int cnt = (N_ - n0) < TILE ? (N_ - n0) : TILE;
    if (tid < cnt) {
      const float sc = score[(size_t)b * N_ + n0 + tid];
      #pragma unroll
      for (int c = 1; c < C_; ++c) {
        float s = s_log[buf][tid * C_ + c] * sc;
        if (bucket_of(s) >= s_thr[c - 1]) {
          const int bc = b * FG + (c - 1);
          int slot = atomicAdd(&cand_cnt[bc], 1);
          if (slot < CAP) {
            cand_s[(size_t)bc * CAP + slot] = s;
            cand_i[(size_t)bc * CAP + slot] = n0 + tid;
          }
        }
      }
    }
    wait_stage_and_sync();
    buf ^= 1;
  }
}

// Per (b,class): bitonic sort (score desc, idx asc), decode top-P boxes,
// greedy NMS, emit P scores/anchors/boxes.
__global__ __launch_bounds__(512) void k_sort_nms(const float* __restrict__ anchors,
                                                  const float* __restrict__ regress,
                                                  const int* __restrict__ cand_cnt,
                                                  const float* __restrict__ cand_s,
                                                  const int* __restrict__ cand_i,
                                                  float* __restrict__ fin_s,
                                                  int* __restrict__ fin_a,
                                                  float* __restrict__ fin_b) {
  __shared__ unsigned long long s_key[CAP];
  __shared__ float s_bx[P_], s_by[P_], s_bz[P_], s_bw[P_], s_sc[P_];
  __shared__ int   s_idx[P_], s_keep[P_], s_cnt;
  const int bc = blockIdx.x, b = bc / FG, tid = threadIdx.x;
  if (tid == 0) {
    int c = cand_cnt[bc];
    s_cnt = c > CAP ? CAP : c;
  }
  __syncthreads();
  const int cnt = s_cnt;
  for (int i = tid; i < CAP; i += 512) {
    unsigned long long k = 0ull;
    if (i < cnt) {
      unsigned sb  = __float_as_uint(cand_s[(size_t)bc * CAP + i]);
      unsigned inv = 0xFFFFFFFFu - (unsigned)cand_i[(size_t)bc * CAP + i];
      k = ((unsigned long long)sb << 32) | inv;
    }
    s_key[i] = k;
  }
  __syncthreads();
  for (unsigned kk = 2; kk <= CAP; kk <<= 1) {
    for (unsigned j = kk >> 1; j > 0; j >>= 1) {
      for (unsigned i = tid; i < CAP; i += 512) {
        unsigned ixj = i ^ j;
        if (ixj > i) {
          unsigned long long a = s_key[i], c = s_key[ixj];
          bool desc = ((i & kk) == 0);
          if (desc ? (a < c) : (a > c)) { s_key[i] = c; s_key[ixj] = a; }
        }
      }
      __syncthreads();
    }
  }
  for (int i = tid; i < P_; i += 512) {
    unsigned long long k = s_key[i];
    float sc = __uint_as_float((unsigned)(k >> 32));
    int idx  = (int)(0xFFFFFFFFu - (unsigned)(k & 0xFFFFFFFFull));
    float4 box = make_float4(0.f, 0.f, 0.f, 0.f);
    if (i < cnt && idx >= 0 && idx < N_) {
      const float4 a = *(const float4*)(anchors + (size_t)idx * 4);
      const float4 r = *(const float4*)(regress + ((size_t)b * N_ + idx) * 4);
      box = decode_box(a, r);
    } else {
      sc = 0.f; idx = 0;
    }
    s_bx[i] = box.x; s_by[i] = box.y; s_bz[i] = box.z; s_bw[i] = box.w;
    s_sc[i] = sc; s_idx[i] = idx;
    s_keep[i] = (sc > SCORE_THR) ? 1 : 0;
  }
  __syncthreads();
  for (int i = 0; i < P_; ++i) {
    if (s_keep[i]) {                         // uniform (LDS) branch
      const float xi = s_bx[i], yi = s_by[i], zi = s_bz[i], wi = s_bw[i];
      const float ai = (zi - xi) * (wi - yi);
      for (int j = i + 1 + tid; j < P_; j += 512) {
        if (s_keep[j]) {
          float xj = s_bx[j], yj = s_by[j], zj = s_bz[j], wj = s_bw[j];
          float aj = (zj - xj) * (wj - yj);
          float iw = fmaxf(fminf(zi, zj) - fmaxf(xi, xj), 0.f);
          float ih = fmaxf(fminf(wi, wj) - fmaxf(yi, yj), 0.f);
          float inter = iw * ih;
          if (inter / (ai + aj - inter + 1e-12f) > IOU_THR) s_keep[j] = 0;
        }
      }
    }
    __syncthreads();
  }
  for (int i = tid; i < P_; i += 512) {
    size_t o = (size_t)bc * P_ + i;          // == [b][cls*P + i]
    fin_s[o] = s_keep[i] ? s_sc[i] : 0.f;
    fin_a[o] = s_idx[i];
    fin_b[o * 4 + 0] = s_bx[i];
    fin_b[o * 4 + 1] = s_by[i];
    fin_b[o * 4 + 2] = s_bz[i];
    fin_b[o * 4 + 3] = s_bw[i];
  }
}

// Per batch: 100x (argmax value, min index) over 10000 LDS scores; gather output.
__global__ __launch_bounds__(1024) void k_topk_out(const float* __restrict__ logits,
                                                   const float* __restrict__ score,
                                                   const float* __restrict__ fin_s,
                                                   const int* __restrict__ fin_a,
                                                   const float* __restrict__ fin_b,
                                                   float* __restrict__ out) {
  __shared__ float s_val[FG * P_];
  __shared__ float s_red[1024];
  __shared__ int   s_arg;
  __shared__ float s_selv[K_];
  __shared__ int   s_seli[K_];
  const int b = blockIdx.x, tid = threadIdx.x;
  for (int i = tid; i < FG * P_; i += 1024) s_val[i] = fin_s[(size_t)b * FG * P_ + i];
  __syncthreads();
  for (int k = 0; k < K_; ++k) {
    float lm = -1.f;
    for (int i = tid; i < FG * P_; i += 1024) lm = fmaxf(lm, s_val[i]);
    s_red[tid] = lm;
    __syncthreads();
    for (int s = 512; s > 0; s >>= 1) {
      if (tid < s) s_red[tid] = fmaxf(s_red[tid], s_red[tid + s]);
      __syncthreads();
    }
    const float gm = s_red[0];
    if (tid == 0) s_arg = 0x7FFFFFFF;
    __syncthreads();
    for (int i = tid; i < FG * P_; i += 1024)
      if (s_val[i] == gm) atomicMin(&s_arg, i);
    __syncthreads();
    if (tid == 0) { int a = s_arg; s_selv[k] = gm; s_seli[k] = a; s_val[a] = -1.f; }
    __syncthreads();
  }
  for (int p = tid; p < K_ * OUTD; p += 1024) {
    const int k = p / OUTD, d = p % OUTD;
    const float v = s_selv[k];
    const int flat = s_seli[k];
    float o = 0.f;
    if (v > 0.f) {
      const size_t base = (size_t)b * FG * P_ + flat;
      if (d < C_) {
        int a = fin_a[base];
        o = logits[((size_t)b * N_ + a) * C_ + d] * score[(size_t)b * N_ + a];
      } else {
        o = fin_b[base * 4 + (d - C_)];
      }
    }
    out[((size_t)b * K_ + k) * OUTD + d] = o;
  }
}

extern "C" void kernel_launch(void* const* d_in, const int* in_sizes, int n_in,
                              void* d_out, int out_size, void* d_ws, size_t ws_size,
                              hipStream_t stream) {
  const float* score   = (const float*)d_in[0];   // (B,N,1)
  const float* logits  = (const float*)d_in[1];   // (B,N,C)
  const float* regress = (const float*)d_in[2];   // (B,N,4)
  const float* anchors = (const float*)d_in[3];   // (N,4)
  float* out = (float*)d_out;

  // Workspace layout (~1.7 MB)
  int*   hist     = (int*)d_ws;                        // B*FG*NB
  int*   cand_cnt = hist + B_ * FG * NB;               // B*FG
  int*   thresh   = cand_cnt + B_ * FG;                // B*FG
  float* cand_s   = (float*)(thresh + B_ * FG);        // B*FG*CAP
  int*   cand_i   = (int*)(cand_s + (size_t)B_ * FG * CAP);
  float* fin_s    = (float*)(cand_i + (size_t)B_ * FG * CAP);   // B*FG*P
  int*   fin_a    = (int*)(fin_s + (size_t)B_ * FG * P_);       // B*FG*P
  float* fin_b    = (float*)(fin_a + (size_t)B_ * FG * P_);     // B*FG*P*4

  const int nzero = B_ * FG * NB + B_ * FG;   // hist + cand_cnt
  k_zero<<<dim3((nzero + 255) / 256), dim3(256), 0, stream>>>(hist, nzero);
  k_hist<<<dim3(32, B_), dim3(256), 0, stream>>>(logits, score, hist);
  k_thresh<<<dim3(B_ * FG), dim3(32), 0, stream>>>(hist, thresh);
  k_compact<<<dim3(32, B_), dim3(256), 0, stream>>>(logits, score, thresh,
                                                    cand_cnt, cand_s, cand_i);
  k_sort_nms<<<dim3(B_ * FG), dim3(512), 0, stream>>>(anchors, regress, cand_cnt,
                                                      cand_s, cand_i, fin_s, fin_a, fin_b);
  k_topk_out<<<dim3(B_), dim3(1024), 0, stream>>>(logits, score, fin_s, fin_a,
                                                  fin_b, out);
}